// NanoMamba_25082609009156
// MI455X (gfx1250) — compile-verified
//
#include <hip/hip_runtime.h>
#include <stdint.h>

// ---------------------------------------------------------------------------
// PCEN (two variants) + spectral-flatness gate blend for (B=64, M=80, T=4000).
//
// MI455X reasoning: ~165MB of unavoidable HBM traffic => ~7us floor at
// 23.3 TB/s; mel (82MB) fits in the 192MB L2 so the second pass is L2-warm.
// The only compute term near that floor is ~8 transcendentals/element, so
// every pow() is exactly one v_log_f32 + one v_exp_f32 (via the amdgcn
// base-2 builtins). The IIR scan along T is parallelized inside a wave32
// with a 5-step Kogge-Stone composition scan (CDNA5 is wave32-only).
// Streaming tiles are double-buffered through LDS with the CDNA5 async data
// mover (global_load_async_to_lds_b32 / s_wait_asynccnt) so loads of tile
// k+1 overlap the serial-carry compute of tile k.
// ---------------------------------------------------------------------------

#define LOG2E 1.4426950408889634f
#define EPSF  1e-6f

// Target builtins: declared by clang itself (no header dependence).
// __builtin_amdgcn_logf  -> v_log_f32  (log2)
// __builtin_amdgcn_exp2f -> v_exp_f32  (exp2)
__device__ __forceinline__ float fexp2_(float x) { return __builtin_amdgcn_exp2f(x); }
__device__ __forceinline__ float flog2_(float x) { return __builtin_amdgcn_logf(x); }

__device__ __forceinline__ float sigmoidf_(float x) {
    return 1.0f / (1.0f + fexp2_(-LOG2E * x));
}
__device__ __forceinline__ float clipf_(float v, float lo, float hi) {
    return fminf(fmaxf(v, lo), hi);
}

// CDNA5 async global->LDS load (ASYNCcnt-tracked). Inline asm: portable
// across ROCm7.2/clang-22 and amdgpu-toolchain/clang-23 (bypasses the
// builtin-arity differences). %0 = per-lane LDS byte address (VDST),
// %1 = 64-bit global address in a VGPR pair, no SADDR ("off" form).
__device__ __forceinline__ void async_load_f32(uint32_t lds_off, const float* gp) {
    uint64_t ga = (uint64_t)(uintptr_t)gp;
    asm volatile("global_load_async_to_lds_b32 %0, %1, off"
                 :: "v"(lds_off), "v"(ga) : "memory");
}
__device__ __forceinline__ void wait_async_le2() {
    asm volatile("s_wait_asynccnt 0x2" ::: "memory");
}
__device__ __forceinline__ void wait_async_0() {
    asm volatile("s_wait_asynccnt 0x0" ::: "memory");
}

// ---------------------------------------------------------------------------
// Kernel 1: gate[b,t] from per-(b,t) reductions over M.
// Threads are consecutive in t -> every m-iteration is a coalesced 128B line.
// ---------------------------------------------------------------------------
__global__ __launch_bounds__(256)
void pcen_gate_kernel(const float* __restrict__ mel,
                      const float* __restrict__ gate_temp,
                      float* __restrict__ gate,
                      int B, int M, int T) {
    int b = blockIdx.x;
    int t = blockIdx.y * blockDim.x + threadIdx.x;
    if (t >= T) return;

    const float* row = mel + (size_t)b * M * T + t;
    int Mlow  = M / 3;        // 26
    int Mhi0  = (2 * M) / 3;  // 53
    float sum = 0.f, sum_log2 = 0.f, lo = 0.f, hi = 0.f;
#pragma unroll 4
    for (int m = 0; m < M; ++m) {
        float x = row[(size_t)m * T];
        sum      += x;
        sum_log2 += flog2_(x + 1e-8f);   // geo mean in base-2: 1 v_log_f32
        if (m < Mlow)  lo += x;
        if (m >= Mhi0) hi += x;
    }
    float invM  = 1.0f / (float)M;
    float geo   = fexp2_(sum_log2 * invM);
    float arith = sum * invM + 1e-8f;
    float sf    = clipf_(geo / arith, 0.f, 1.f);
    float lowm  = lo / (float)Mlow;
    float highm = hi / (float)(M - Mhi0);
    float tilt  = clipf_(lowm / (lowm + highm + 1e-8f), 0.f, 1.f);
    float sfadj = sf + (1.f - sf) * fmaxf(tilt - 0.6f, 0.f);
    gate[(size_t)b * T + t] = sigmoidf_(gate_temp[0] * (sfadj - 0.5f));
}

// ---------------------------------------------------------------------------
// Kernel 2: one wave32 per (b,m) row. Dual IIR scan + PCEN + blend.
// ---------------------------------------------------------------------------
#define WAVES_PB 8
#define TILE     32

__global__ __launch_bounds__(256)
void pcen_scan_kernel(const float* __restrict__ mel,
                      const float* __restrict__ ls_ns, const float* __restrict__ la_ns,
                      const float* __restrict__ ld_ns, const float* __restrict__ lr_ns,
                      const float* __restrict__ ls_st, const float* __restrict__ la_st,
                      const float* __restrict__ ld_st, const float* __restrict__ lr_st,
                      const float* __restrict__ gate,
                      float* __restrict__ out,
                      int B, int M, int T) {
    // per-wave double buffer: [buf][0:32]=mel tile, [buf][32:64]=gate tile
    __shared__ float lds[WAVES_PB][2][2 * TILE];

    int lane = threadIdx.x & 31;
    int wv   = threadIdx.x >> 5;
    int row  = blockIdx.x * WAVES_PB + wv;
    if (row >= B * M) return;                 // uniform per wave
    int b = row / M, m = row - b * M;

    // Per-channel parameters (uniform across the wave -> scalar-cached loads).
    float s_ns = clipf_(sigmoidf_(ls_ns[m]), 0.05f, 0.3f);
    float a_ns = clipf_(sigmoidf_(la_ns[m]), 0.9f, 0.999f);
    float d_ns = clipf_(fexp2_(LOG2E * ld_ns[m]), 0.5f, 5.0f);
    float r_ns = clipf_(sigmoidf_(lr_ns[m]), 0.05f, 0.25f);
    float s_st = clipf_(sigmoidf_(ls_st[m]), 0.05f, 0.3f);
    float a_st = clipf_(sigmoidf_(la_st[m]), 0.9f, 0.999f);
    float d_st = clipf_(fexp2_(LOG2E * ld_st[m]), 0.001f, 0.1f);
    float r_st = clipf_(sigmoidf_(lr_st[m]), 0.05f, 0.25f);
    float c_ns = 1.f - s_ns, c_st = 1.f - s_st;
    float dr_ns = fexp2_(r_ns * flog2_(d_ns));
    float dr_st = fexp2_(r_st * flog2_(d_st));

    const float* mrow = mel  + (size_t)row * T;
    const float* grow = gate + (size_t)b * T;
    float*       orow = out  + (size_t)row * T;

    // jax.lax.scan(init = x[0]) => sm[0] = (1-s)*x0 + s*x0 = x0, so carry = x0.
    float carry_ns = mrow[0];
    float carry_st = carry_ns;

    // LDS byte address of this wave's staging area. On AMDGPU the low 32 bits
    // of a flat address into the LDS aperture are the DS byte offset the
    // async instruction wants in VDST.
    uint32_t lbase = (uint32_t)(uintptr_t)(&lds[wv][0][0]);
    const uint32_t BUFSTRIDE = 2u * TILE * 4u;   // 256 bytes

    int ntiles = T / TILE;                       // 125 (T=4000, exact)

    // Prefetch tile 0 (2 async ops: mel + gate).
    async_load_f32(lbase + (uint32_t)lane * 4u,          mrow + lane);
    async_load_f32(lbase + (uint32_t)(TILE + lane) * 4u, grow + lane);

    for (int k = 0; k < ntiles; ++k) {
        int cb = k & 1;
        if (k + 1 < ntiles) {
            // Issue tile k+1 into the other buffer, then wait until only those
            // 2 ops remain outstanding (async loads complete in order).
            uint32_t off = lbase + (uint32_t)((k + 1) & 1) * BUFSTRIDE;
            int t1 = (k + 1) * TILE;
            async_load_f32(off + (uint32_t)lane * 4u,          mrow + t1 + lane);
            async_load_f32(off + (uint32_t)(TILE + lane) * 4u, grow + t1 + lane);
            wait_async_le2();
        } else {
            wait_async_0();
        }

        float x = lds[wv][cb][lane];
        float g = lds[wv][cb][TILE + lane];

        // Kogge-Stone scan of y <- A*y + B composition, both variants at once.
        float A1 = c_ns, B1 = s_ns * x;
        float A2 = c_st, B2 = s_st * x;
#pragma unroll
        for (int off2 = 1; off2 < 32; off2 <<= 1) {
            float a1 = __shfl_up(A1, off2, 32);
            float b1 = __shfl_up(B1, off2, 32);
            float a2 = __shfl_up(A2, off2, 32);
            float b2 = __shfl_up(B2, off2, 32);
            if (lane >= off2) {
                B1 = fmaf(A1, b1, B1); A1 *= a1;
                B2 = fmaf(A2, b2, B2); A2 *= a2;
            }
        }
        float sm_ns = fmaf(A1, carry_ns, B1);
        float sm_st = fmaf(A2, carry_st, B2);
        carry_ns = __shfl(sm_ns, 31, 32);
        carry_st = __shfl(sm_st, 31, 32);

        // (eps+sm)^(-alpha), (x*gain+d)^r - d^r : 2x v_log_f32 + 2x v_exp_f32 each.
        float gain_ns = fexp2_(-a_ns * flog2_(EPSF + sm_ns));
        float o_ns    = fexp2_(r_ns * flog2_(fmaf(x, gain_ns, d_ns))) - dr_ns;
        float gain_st = fexp2_(-a_st * flog2_(EPSF + sm_st));
        float o_st    = fexp2_(r_st * flog2_(fmaf(x, gain_st, d_st))) - dr_st;

        orow[k * TILE + lane] = fmaf(g, o_st - o_ns, o_ns);
    }
}

// ---------------------------------------------------------------------------
extern "C" void kernel_launch(void* const* d_in, const int* in_sizes, int n_in,
                              void* d_out, int out_size, void* d_ws, size_t ws_size,
                              hipStream_t stream) {
    const float* mel   = (const float*)d_in[0];
    const float* lsns  = (const float*)d_in[1];
    const float* lans  = (const float*)d_in[2];
    const float* ldns  = (const float*)d_in[3];
    const float* lrns  = (const float*)d_in[4];
    const float* lsst  = (const float*)d_in[5];
    const float* last  = (const float*)d_in[6];
    const float* ldst  = (const float*)d_in[7];
    const float* lrst  = (const float*)d_in[8];
    const float* gtemp = (const float*)d_in[9];

    int M = in_sizes[1];                 // 80
    int B = 64;
    int T = in_sizes[0] / (B * M);       // 4000

    float* gate = (float*)d_ws;          // B*T floats = 1 MB scratch
    float* outp = (float*)d_out;

    dim3 g1((unsigned)B, (unsigned)((T + 255) / 256));
    pcen_gate_kernel<<<g1, 256, 0, stream>>>(mel, gtemp, gate, B, M, T);

    int rows   = B * M;                  // 5120 waves
    int blocks = (rows + WAVES_PB - 1) / WAVES_PB;
    pcen_scan_kernel<<<blocks, 256, 0, stream>>>(mel,
        lsns, lans, ldns, lrns, lsst, last, ldst, lrst,
        gate, outp, B, M, T);
}